// BaseLayer_1864015807157
// MI455X (gfx1250) — compile-verified
//
#include <hip/hip_runtime.h>
#include <math.h>

#define S_TOK 8192
#define MDIM  1024
#define EXP   4
#define HDIM  4096
#define ODIM  1024
#define CAP   2048

typedef float v2f __attribute__((ext_vector_type(2)));
typedef float v8f __attribute__((ext_vector_type(8)));
typedef int   vsi4 __attribute__((__vector_size__(4 * sizeof(int))));

// ---- CDNA5 async global->LDS path (guarded; falls back to sync copy) ------
#if defined(__has_builtin)
#if __has_builtin(__builtin_amdgcn_global_load_async_to_lds_b128)
#define HAVE_ASYNC_LDS 1
#endif
#endif

__device__ __forceinline__ void copy16_to_lds(float* lds_dst, const float* gsrc)
{
#ifdef HAVE_ASYNC_LDS
    __builtin_amdgcn_global_load_async_to_lds_b128(
        (__attribute__((address_space(1))) vsi4*)gsrc,
        (__attribute__((address_space(3))) vsi4*)lds_dst, 0, 0);
#else
    *(float4*)lds_dst = *(const float4*)gsrc;
#endif
}

__device__ __forceinline__ void wait_async_lds()
{
#ifdef HAVE_ASYNC_LDS
#if __has_builtin(__builtin_amdgcn_s_wait_asynccnt)
    __builtin_amdgcn_s_wait_asynccnt(0);
#else
    asm volatile("s_wait_asynccnt 0x0" ::: "memory");
#endif
#endif
}

// ---------------------------------------------------------------------------
// Gate: one wave32 per token. fp32 dot with gate_w (rows are float4 since E=4),
// shuffle-reduce, softmax + first-max argmax on lane 0.
// ---------------------------------------------------------------------------
__global__ __launch_bounds__(256)
void moe_gate(const float* __restrict__ feat, const float* __restrict__ gw,
              int* __restrict__ token_idx, float* __restrict__ token_gate,
              float* __restrict__ probs)
{
    int wave = threadIdx.x >> 5;
    int lane = threadIdx.x & 31;
    int s = blockIdx.x * 8 + wave;
    const float4* gw4 = (const float4*)gw;
    const float* f = feat + (size_t)s * MDIM;

    float a0 = 0.f, a1 = 0.f, a2 = 0.f, a3 = 0.f;
    for (int m = lane; m < MDIM; m += 32) {
        float x = f[m];
        float4 w = gw4[m];
        a0 = fmaf(x, w.x, a0);
        a1 = fmaf(x, w.y, a1);
        a2 = fmaf(x, w.z, a2);
        a3 = fmaf(x, w.w, a3);
    }
    for (int off = 16; off > 0; off >>= 1) {
        a0 += __shfl_xor(a0, off, 32);
        a1 += __shfl_xor(a1, off, 32);
        a2 += __shfl_xor(a2, off, 32);
        a3 += __shfl_xor(a3, off, 32);
    }
    if (lane == 0) {
        float l[4] = {a0, a1, a2, a3};
        float mx = l[0];
        int arg = 0;
        for (int e = 1; e < 4; ++e)
            if (l[e] > mx) { mx = l[e]; arg = e; }   // first max, like jnp.argmax
        float p[4];
        float sum = 0.f;
        for (int e = 0; e < 4; ++e) { p[e] = expf(l[e] - mx); sum += p[e]; }
        float inv = 1.f / sum;
        for (int e = 0; e < 4; ++e) { p[e] *= inv; probs[s * 4 + e] = p[e]; }
        token_idx[s]  = arg;
        token_gate[s] = p[arg];
    }
}

// ---------------------------------------------------------------------------
// l_aux = E * sum_e mean(gates[:,e]) * mean(onehot_argmax[:,e]).
// Single block, fixed-order accumulation + tree reduce => deterministic.
// ---------------------------------------------------------------------------
__global__ __launch_bounds__(256)
void moe_laux(const float* __restrict__ probs, const int* __restrict__ token_idx,
              float* __restrict__ laux_out)
{
    __shared__ float red[256][8];
    int tid = threadIdx.x;
    float me[4] = {0.f, 0.f, 0.f, 0.f};
    float ce[4] = {0.f, 0.f, 0.f, 0.f};
    for (int s = tid; s < S_TOK; s += 256) {
        me[0] += probs[s * 4 + 0];
        me[1] += probs[s * 4 + 1];
        me[2] += probs[s * 4 + 2];
        me[3] += probs[s * 4 + 3];
        ce[token_idx[s]] += 1.f;
    }
    for (int e = 0; e < 4; ++e) { red[tid][e] = me[e]; red[tid][4 + e] = ce[e]; }
    __syncthreads();
    for (int off = 128; off > 0; off >>= 1) {
        if (tid < off)
            for (int k = 0; k < 8; ++k) red[tid][k] += red[tid + off][k];
        __syncthreads();
    }
    if (tid == 0) {
        float l = 0.f;
        for (int e = 0; e < 4; ++e)
            l += (red[0][e] / (float)S_TOK) * (red[0][4 + e] / (float)S_TOK);
        laux_out[0] = l * (float)EXP;
    }
}

// ---------------------------------------------------------------------------
// Exact token-order capacity assignment: prefix count of argmax one-hots.
// 1024 threads x 8 contiguous tokens + Hillis-Steele scan of 4 counters.
// Builds reverse map slot_token[e*CAP + loc] = token (or -1 if empty).
// ---------------------------------------------------------------------------
__global__ __launch_bounds__(1024)
void moe_scan(const int* __restrict__ token_idx, int* __restrict__ slot_token)
{
    __shared__ int sc[1024 * 4];
    int t = threadIdx.x;
    for (int i = t; i < EXP * CAP; i += 1024) slot_token[i] = -1;

    int le[8];
    int c[4] = {0, 0, 0, 0};
    int base = t * 8;
    for (int k = 0; k < 8; ++k) {
        int e = token_idx[base + k];
        le[k] = e;
        c[e]++;
    }
    for (int e = 0; e < 4; ++e) sc[t * 4 + e] = c[e];
    __syncthreads();
    for (int off = 1; off < 1024; off <<= 1) {
        int v[4], u[4] = {0, 0, 0, 0};
        for (int e = 0; e < 4; ++e) v[e] = sc[t * 4 + e];
        if (t >= off)
            for (int e = 0; e < 4; ++e) u[e] = sc[(t - off) * 4 + e];
        __syncthreads();
        for (int e = 0; e < 4; ++e) sc[t * 4 + e] = v[e] + u[e];
        __syncthreads();
    }
    int p[4];
    for (int e = 0; e < 4; ++e) p[e] = sc[t * 4 + e] - c[e];  // exclusive prefix
    for (int k = 0; k < 8; ++k) {
        int e = le[k];
        int loc = p[e]++;
        if (loc < CAP) slot_token[e * CAP + loc] = base + k;  // overflow = dropped
    }
}

__device__ __forceinline__ float gelu_tanh(float x)
{
    // jax.nn.gelu default (approximate=True)
    return 0.5f * x * (1.f + tanhf(0.7978845608028654f * (x + 0.044715f * x * x * x)));
}

// ---------------------------------------------------------------------------
// fc1: gather(A rows via slot_token) @ fc1_w[e] + b1 -> gelu -> hbuf.
// Double-buffered LDS, async global->LDS copies overlapped with WMMA.
// Block = 128x128 tile, 8 waves, each wave 64x32 via 4x2 wmma_f32_16x16x4_f32.
// ---------------------------------------------------------------------------
__global__ __launch_bounds__(256)
void moe_fc1(const float* __restrict__ feat, const float* __restrict__ w1,
             const float* __restrict__ b1, const int* __restrict__ slot_token,
             float* __restrict__ hbuf)
{
    __shared__ float A_lds[2][128][20];   // 16B-aligned rows (80B stride)
    __shared__ float B_lds[2][16][132];   // 16B-aligned rows (528B stride)
    __shared__ int   row_tok[128];

    int e  = blockIdx.z;
    int c0 = blockIdx.y * 128;
    int n0 = blockIdx.x * 128;
    int tid = threadIdx.x;

    if (tid < 128) row_tok[tid] = slot_token[e * CAP + c0 + tid];
    __syncthreads();

    int wave = tid >> 5, lane = tid & 31;
    int wr = wave >> 2, wc = wave & 3;          // 2x4 wave grid
    int m  = lane & 15, ko = (lane >> 4) * 2;   // ISA A/B fragment lane layout

    v8f acc[4][2];
    for (int i = 0; i < 4; ++i)
        for (int j = 0; j < 2; ++j)
            for (int r = 0; r < 8; ++r) acc[i][j][r] = 0.f;

    const float* wptr = w1 + (size_t)e * MDIM * HDIM + n0;

    auto issue = [&](int k0, int buf) {
        #pragma unroll
        for (int it = 0; it < 2; ++it) {       // A tile: 512 float4 slots
            int id = tid + it * 256;
            int r = id >> 2, kq = id & 3;
            int tk = row_tok[r];
            float* dst = &A_lds[buf][r][kq * 4];
            if (tk >= 0)
                copy16_to_lds(dst, feat + (size_t)tk * MDIM + k0 + kq * 4);
            else
                *(float4*)dst = make_float4(0.f, 0.f, 0.f, 0.f);
        }
        #pragma unroll
        for (int it = 0; it < 2; ++it) {       // B tile: 512 float4 slots
            int id = tid + it * 256;
            int kk = id >> 5, cq = id & 31;
            copy16_to_lds(&B_lds[buf][kk][cq * 4],
                          wptr + (size_t)(k0 + kk) * HDIM + cq * 4);
        }
    };

    const int NST = MDIM / 16;
    issue(0, 0);
    for (int st = 0; st < NST; ++st) {
        wait_async_lds();
        __syncthreads();                        // stage st resident for all waves
        if (st + 1 < NST) issue((st + 1) * 16, (st + 1) & 1);  // overlap w/ compute
        int buf = st & 1;

        #pragma unroll
        for (int kk = 0; kk < 16; kk += 4) {
            v2f af[4], bf[2];
            #pragma unroll
            for (int i = 0; i < 4; ++i) {
                int rr = wr * 64 + i * 16 + m;
                af[i][0] = A_lds[buf][rr][kk + ko];
                af[i][1] = A_lds[buf][rr][kk + ko + 1];
            }
            #pragma unroll
            for (int j = 0; j < 2; ++j) {
                int cc = wc * 32 + j * 16 + m;
                bf[j][0] = B_lds[buf][kk + ko][cc];
                bf[j][1] = B_lds[buf][kk + ko + 1][cc];
            }
            #pragma unroll
            for (int i = 0; i < 4; ++i)
                #pragma unroll
                for (int j = 0; j < 2; ++j)
                    acc[i][j] = __builtin_amdgcn_wmma_f32_16x16x4_f32(
                        false, af[i], false, bf[j], (short)0, acc[i][j], false, false);
        }
    }

    int hi = lane >> 4;
    #pragma unroll
    for (int i = 0; i < 4; ++i)
        #pragma unroll
        for (int j = 0; j < 2; ++j) {
            int col = n0 + wc * 32 + j * 16 + m;
            float bias = b1[e * HDIM + col];
            #pragma unroll
            for (int r = 0; r < 8; ++r) {
                int row = c0 + wr * 64 + i * 16 + hi * 8 + r;
                hbuf[((size_t)e * CAP + row) * HDIM + col] =
                    gelu_tanh(acc[i][j][r] + bias);
            }
        }
}

// ---------------------------------------------------------------------------
// fc2 + combine: hbuf[e] @ fc2_w[e] + b2, scaled by gate, scattered to the
// owning token's output row (d_out pre-zeroed; dropped/empty slots skipped).
// Same double-buffered async pipeline.
// ---------------------------------------------------------------------------
__global__ __launch_bounds__(256)
void moe_fc2(const float* __restrict__ hbuf, const float* __restrict__ w2,
             const float* __restrict__ b2, const int* __restrict__ slot_token,
             const float* __restrict__ token_gate, float* __restrict__ out)
{
    __shared__ float A_lds[2][128][20];
    __shared__ float B_lds[2][16][132];
    __shared__ int   row_tok[128];
    __shared__ float row_gate[128];

    int e  = blockIdx.z;
    int c0 = blockIdx.y * 128;
    int n0 = blockIdx.x * 128;
    int tid = threadIdx.x;

    if (tid < 128) {
        int tk = slot_token[e * CAP + c0 + tid];
        row_tok[tid]  = tk;
        row_gate[tid] = (tk >= 0) ? token_gate[tk] : 0.f;
    }
    __syncthreads();

    int wave = tid >> 5, lane = tid & 31;
    int wr = wave >> 2, wc = wave & 3;
    int m  = lane & 15, ko = (lane >> 4) * 2;

    v8f acc[4][2];
    for (int i = 0; i < 4; ++i)
        for (int j = 0; j < 2; ++j)
            for (int r = 0; r < 8; ++r) acc[i][j][r] = 0.f;

    const float* aptr = hbuf + ((size_t)e * CAP + c0) * HDIM;
    const float* wptr = w2 + (size_t)e * HDIM * ODIM + n0;

    auto issue = [&](int k0, int buf) {
        #pragma unroll
        for (int it = 0; it < 2; ++it) {
            int id = tid + it * 256;
            int r = id >> 2, kq = id & 3;
            copy16_to_lds(&A_lds[buf][r][kq * 4],
                          aptr + (size_t)r * HDIM + k0 + kq * 4);
        }
        #pragma unroll
        for (int it = 0; it < 2; ++it) {
            int id = tid + it * 256;
            int kk = id >> 5, cq = id & 31;
            copy16_to_lds(&B_lds[buf][kk][cq * 4],
                          wptr + (size_t)(k0 + kk) * ODIM + cq * 4);
        }
    };

    const int NST = HDIM / 16;
    issue(0, 0);
    for (int st = 0; st < NST; ++st) {
        wait_async_lds();
        __syncthreads();
        if (st + 1 < NST) issue((st + 1) * 16, (st + 1) & 1);
        int buf = st & 1;

        #pragma unroll
        for (int kk = 0; kk < 16; kk += 4) {
            v2f af[4], bf[2];
            #pragma unroll
            for (int i = 0; i < 4; ++i) {
                int rr = wr * 64 + i * 16 + m;
                af[i][0] = A_lds[buf][rr][kk + ko];
                af[i][1] = A_lds[buf][rr][kk + ko + 1];
            }
            #pragma unroll
            for (int j = 0; j < 2; ++j) {
                int cc = wc * 32 + j * 16 + m;
                bf[j][0] = B_lds[buf][kk + ko][cc];
                bf[j][1] = B_lds[buf][kk + ko + 1][cc];
            }
            #pragma unroll
            for (int i = 0; i < 4; ++i)
                #pragma unroll
                for (int j = 0; j < 2; ++j)
                    acc[i][j] = __builtin_amdgcn_wmma_f32_16x16x4_f32(
                        false, af[i], false, bf[j], (short)0, acc[i][j], false, false);
        }
    }

    int hi = lane >> 4;
    #pragma unroll
    for (int i = 0; i < 4; ++i)
        #pragma unroll
        for (int j = 0; j < 2; ++j) {
            int col = n0 + wc * 32 + j * 16 + m;
            float bias = b2[e * ODIM + col];
            #pragma unroll
            for (int r = 0; r < 8; ++r) {
                int rl = wr * 64 + i * 16 + hi * 8 + r;
                int tk = row_tok[rl];
                if (tk >= 0)
                    out[(size_t)tk * ODIM + col] =
                        row_gate[rl] * (acc[i][j][r] + bias);
            }
        }
}

// ---------------------------------------------------------------------------
// Workspace layout (bytes):
//   0        token_idx  int[8192]
//   32768    token_gate float[8192]
//   65536    slot_token int[8192]
//   98304    probs      float[8192*4]
//   229376   hbuf       float[4*2048*4096]   (128 MB)
// ---------------------------------------------------------------------------
extern "C" void kernel_launch(void* const* d_in, const int* in_sizes, int n_in,
                              void* d_out, int out_size, void* d_ws, size_t ws_size,
                              hipStream_t stream)
{
    (void)in_sizes; (void)n_in; (void)out_size; (void)ws_size;

    const float* hidden = (const float*)d_in[0];
    const float* gate_w = (const float*)d_in[1];
    const float* fc1_w  = (const float*)d_in[2];
    const float* fc1_b  = (const float*)d_in[3];
    const float* fc2_w  = (const float*)d_in[4];
    const float* fc2_b  = (const float*)d_in[5];
    float* out = (float*)d_out;

    char* ws = (char*)d_ws;
    int*   token_idx  = (int*)(ws);
    float* token_gate = (float*)(ws + 32768);
    int*   slot_token = (int*)(ws + 65536);
    float* probs      = (float*)(ws + 98304);
    float* hbuf       = (float*)(ws + 229376);

    (void)hipMemsetAsync(d_out, 0, (size_t)S_TOK * ODIM * sizeof(float), stream);

    moe_gate<<<S_TOK / 8, 256, 0, stream>>>(hidden, gate_w, token_idx, token_gate, probs);
    moe_laux<<<1, 256, 0, stream>>>(probs, token_idx, out + (size_t)S_TOK * ODIM);
    moe_scan<<<1, 1024, 0, stream>>>(token_idx, slot_token);
    moe_fc1<<<dim3(HDIM / 128, CAP / 128, EXP), 256, 0, stream>>>(
        hidden, fc1_w, fc1_b, slot_token, hbuf);
    moe_fc2<<<dim3(ODIM / 128, CAP / 128, EXP), 256, 0, stream>>>(
        hbuf, fc2_w, fc2_b, slot_token, token_gate, out);
}